// PetaloMixer_44195213476097
// MI455X (gfx1250) — compile-verified
//
#include <hip/hip_runtime.h>
#include <hip/hip_bf16.h>

// ---- problem constants (from reference) ----
#define CN   256   // channels
#define OUTN 256   // output channels
#define DMc  64    // C/4
#define DSc  16    // state size
#define DCc  4     // conv kernel
#define DIc  128   // inner dim (2*DM)
#define DTRc 4     // dt rank
#define Bc   2
#define Lc   4096  // Dd*H*W = 8*16*32
#define NTc  (Bc*Lc)   // 8192 tokens
#define Pc   4     // parts

typedef float v2f __attribute__((ext_vector_type(2)));
typedef float v8f __attribute__((ext_vector_type(8)));

__device__ __forceinline__ float silu_f(float x) {
    return x / (1.0f + __expf(-x));
}

// -------------------------------------------------------------------------
// Kernel 1: LayerNorm over C with transpose (B,C,L) -> token-major (NT,C)
// 32 tokens per block, LDS tile for coalesced global reads.
// -------------------------------------------------------------------------
__global__ __launch_bounds__(256) void ln1_kernel(
    const float* __restrict__ x, const float* __restrict__ g,
    const float* __restrict__ be, float* __restrict__ xn)
{
    __shared__ float tile[CN][33];   // [channel][token], padded
    __shared__ float redS[256], redQ[256];
    __shared__ float muA[32], riA[32];

    int bId  = blockIdx.x;                  // 0 .. B*L/32-1
    int bat  = bId / (Lc / 32);
    int l0   = (bId % (Lc / 32)) * 32;
    int tid  = threadIdx.x;

    int col  = tid & 31;       // token in tile
    int rq   = tid >> 5;       // 0..7
    for (int c0 = 0; c0 < CN; c0 += 8) {
        int c = c0 + rq;
        tile[c][col] = x[((size_t)bat * CN + c) * Lc + l0 + col];
    }
    __syncthreads();

    // 8 threads per token accumulate partial stats over 32 channels each
    int j = tid >> 3;          // token 0..31
    int q = tid & 7;
    float s = 0.f, sq = 0.f;
    #pragma unroll 4
    for (int cc = 0; cc < 32; ++cc) {
        float v = tile[q * 32 + cc][j];
        s += v; sq += v * v;
    }
    redS[tid] = s; redQ[tid] = sq;
    __syncthreads();
    if (q == 0) {
        float ts = 0.f, tq = 0.f;
        #pragma unroll
        for (int k = 0; k < 8; ++k) { ts += redS[j * 8 + k]; tq += redQ[j * 8 + k]; }
        float mu  = ts * (1.0f / CN);
        float var = tq * (1.0f / CN) - mu * mu;
        muA[j] = mu; riA[j] = rsqrtf(var + 1e-5f);
    }
    __syncthreads();

    float gc = g[tid];
    float bc_ = be[tid];
    for (int jj = 0; jj < 32; ++jj) {
        float v = (tile[tid][jj] - muA[jj]) * riA[jj] * gc + bc_;
        xn[((size_t)bat * Lc + l0 + jj) * CN + tid] = v;
    }
}

// -------------------------------------------------------------------------
// Kernel 2: in_proj, WMMA f32.  xz[p] = xn[:,p*64:(p+1)*64] @ W^T (256 out)
// Block = 128 threads = 4 waves; tile = 16 tokens x 256 outputs.
// -------------------------------------------------------------------------
__global__ __launch_bounds__(128) void inproj_kernel(
    const float* __restrict__ xn, const float* __restrict__ w,
    float* __restrict__ xz)
{
    int p    = blockIdx.y;
    int t0   = blockIdx.x * 16;
    int wv   = threadIdx.x >> 5;
    int lane = threadIdx.x & 31;
    int row  = lane & 15;      // m for A, n for B/D
    int kh   = lane >> 4;      // k-half selector

    const float* apt = xn + ((size_t)t0 + row) * CN + p * DMc;
    v8f acc[4] = {};
    for (int kk = 0; kk < DMc; kk += 4) {
        v2f a; a[0] = apt[kk + 2 * kh]; a[1] = apt[kk + 2 * kh + 1];
        #pragma unroll
        for (int nt = 0; nt < 4; ++nt) {
            int n0 = wv * 64 + nt * 16;
            const float* bpt = w + ((size_t)n0 + row) * DMc + kk + 2 * kh;
            v2f b; b[0] = bpt[0]; b[1] = bpt[1];
            acc[nt] = __builtin_amdgcn_wmma_f32_16x16x4_f32(
                false, a, false, b, (short)0, acc[nt], false, false);
        }
    }
    #pragma unroll
    for (int nt = 0; nt < 4; ++nt) {
        int n0 = wv * 64 + nt * 16;
        #pragma unroll
        for (int r = 0; r < 8; ++r) {
            int m = r + 8 * kh;
            xz[(size_t)p * NTc * 256 + ((size_t)t0 + m) * 256 + n0 + row] = acc[nt][r];
        }
    }
}

// -------------------------------------------------------------------------
// Kernel 3: depthwise causal conv(4) + SiLU over xz cols 0..127 -> xc
// -------------------------------------------------------------------------
__global__ __launch_bounds__(128) void conv_kernel(
    const float* __restrict__ xz, const float* __restrict__ cw,
    const float* __restrict__ cb, float* __restrict__ xc)
{
    int p = blockIdx.y;
    size_t t = blockIdx.x;
    int l  = (int)(t % Lc);
    int ch = threadIdx.x;
    size_t base = (size_t)p * NTc * 256;
    float s = cb[ch];
    #pragma unroll
    for (int k = 0; k < DCc; ++k) {
        int ll = l - (DCc - 1) + k;
        if (ll >= 0)
            s += xz[base + (t - (DCc - 1) + k) * 256 + ch] * cw[ch * DCc + k];
    }
    xc[((size_t)p * NTc + t) * DIc + ch] = silu_f(s);
}

// -------------------------------------------------------------------------
// Kernel 4: x_proj (36 outs) + dt_proj + softplus.  One block per token.
// -------------------------------------------------------------------------
__global__ __launch_bounds__(128) void xproj_kernel(
    const float* __restrict__ xc, const float* __restrict__ xpw,
    const float* __restrict__ dtw, const float* __restrict__ dtb,
    float* __restrict__ dtv, float* __restrict__ bcb)
{
    __shared__ float xs[DIc];
    __shared__ float dbl[DTRc + 2 * DSc];   // 36
    int p = blockIdx.y;
    size_t t = blockIdx.x;
    int tid = threadIdx.x;
    size_t base = (size_t)p * NTc + t;

    xs[tid] = xc[base * DIc + tid];
    __syncthreads();
    if (tid < DTRc + 2 * DSc) {
        const float* wr = xpw + tid * DIc;
        float s = 0.f;
        #pragma unroll 4
        for (int ch = 0; ch < DIc; ++ch) s += wr[ch] * xs[ch];
        dbl[tid] = s;
    }
    __syncthreads();

    float s = dtb[tid];
    #pragma unroll
    for (int r = 0; r < DTRc; ++r) s += dbl[r] * dtw[tid * DTRc + r];
    float sp = (s > 20.f) ? s : log1pf(__expf(s));
    dtv[base * DIc + tid] = sp;
    if (tid < 2 * DSc) bcb[base * 32 + tid] = dbl[DTRc + tid];
}

// -------------------------------------------------------------------------
// Kernel 5: selective scan. One block per (batch, part); thread = d channel.
// h[16] lives in registers. B/C chunks staged into LDS with the CDNA5
// async-copy path: GLOBAL_LOAD_ASYNC_TO_LDS_B128 (ASYNCcnt) instead of a
// VGPR round-trip; s_wait_asynccnt 0 before the barrier. Fuses +D*xc and
// *silu(z). Writes y into xz columns 0..127 (dead xc_raw columns).
// -------------------------------------------------------------------------
__global__ __launch_bounds__(128) void scan_kernel(
    const float* __restrict__ dtv, const float* __restrict__ bcb,
    const float* __restrict__ xc, const float* __restrict__ A_log,
    const float* __restrict__ Dp, float* __restrict__ xz)
{
    __shared__ float bcS[64 * 32];   // [tt][j]: j<16 -> B, j>=16 -> C
    int p = blockIdx.x & 3;
    int bat = blockIdx.x >> 2;
    int d = threadIdx.x;

    float a[DSc];
    #pragma unroll
    for (int s = 0; s < DSc; ++s) a[s] = -__expf(A_log[d * DSc + s]);
    float Dd = Dp[d];
    float h[DSc];
    #pragma unroll
    for (int s = 0; s < DSc; ++s) h[s] = 0.f;

    // raw LDS byte offset of the staging tile (aperture addr low 32 bits)
    unsigned ldsbase = (unsigned)(uintptr_t)(void*)bcS;

    for (int c0 = 0; c0 < Lc; c0 += 64) {
        __syncthreads();   // previous chunk fully consumed
        // async-copy 64x32 floats (8 KB) contiguously: 4 x b128 per lane
        const char* gsrc = (const char*)(bcb +
            ((size_t)p * NTc + (size_t)bat * Lc + c0) * 32);
        #pragma unroll
        for (int k = 0; k < 4; ++k) {
            unsigned off = ((unsigned)threadIdx.x + k * 128u) * 16u;
            unsigned lds_addr = ldsbase + off;
            const void* gaddr = gsrc + off;
            asm volatile("global_load_async_to_lds_b128 %0, %1, off"
                         :: "v"(lds_addr), "v"(gaddr) : "memory");
        }
        asm volatile("s_wait_asynccnt 0" ::: "memory");
        __syncthreads();

        for (int tt = 0; tt < 64; ++tt) {
            size_t base = (size_t)p * NTc + (size_t)bat * Lc + c0 + tt;
            float dt  = dtv[base * DIc + d];
            float xcv = xc[base * DIc + d];
            float z   = xz[base * 256 + DIc + d];
            float dx  = dt * xcv;
            float y = 0.f;
            #pragma unroll
            for (int s = 0; s < DSc; ++s) {
                float e = __expf(dt * a[s]);
                h[s] = e * h[s] + dx * bcS[tt * 32 + s];
                y += h[s] * bcS[tt * 32 + 16 + s];
            }
            xz[base * 256 + d] = (y + Dd * xcv) * silu_f(z);
        }
    }
}

// -------------------------------------------------------------------------
// Kernel 6: out_proj, WMMA f32.  xm[:,p*64+n] = y @ W^T + skip*xn_part
// Block = 128 threads = 4 waves; tile = 16 tokens x 64 outputs.
// -------------------------------------------------------------------------
__global__ __launch_bounds__(128) void outproj_kernel(
    const float* __restrict__ ybuf, const float* __restrict__ w,
    const float* __restrict__ xn, const float* __restrict__ sk,
    float* __restrict__ xm)
{
    int p    = blockIdx.y;
    int t0   = blockIdx.x * 16;
    int wv   = threadIdx.x >> 5;
    int lane = threadIdx.x & 31;
    int row  = lane & 15;
    int kh   = lane >> 4;
    int n0   = wv * 16;

    const float* apt = ybuf + ((size_t)p * NTc + t0 + row) * 256;   // y in cols 0..127
    const float* bpt = w + ((size_t)n0 + row) * DIc;
    v8f acc = {};
    for (int kk = 0; kk < DIc; kk += 4) {
        v2f a, b;
        a[0] = apt[kk + 2 * kh]; a[1] = apt[kk + 2 * kh + 1];
        b[0] = bpt[kk + 2 * kh]; b[1] = bpt[kk + 2 * kh + 1];
        acc = __builtin_amdgcn_wmma_f32_16x16x4_f32(
            false, a, false, b, (short)0, acc, false, false);
    }
    float skip = sk[0];
    #pragma unroll
    for (int r = 0; r < 8; ++r) {
        int m = r + 8 * kh;
        size_t tt = (size_t)t0 + m;
        int c = p * DMc + n0 + row;
        xm[tt * CN + c] = acc[r] + skip * xn[tt * CN + c];
    }
}

// -------------------------------------------------------------------------
// Kernel 7: LayerNorm on token-major data (wave per token, wave32 shuffles)
// -------------------------------------------------------------------------
__global__ __launch_bounds__(256) void ln2_kernel(
    const float* __restrict__ xm, const float* __restrict__ g,
    const float* __restrict__ be, float* __restrict__ out)
{
    int wv = threadIdx.x >> 5, lane = threadIdx.x & 31;
    size_t t = (size_t)blockIdx.x * 8 + wv;
    const float* row = xm + t * CN;
    float v[8]; float s = 0.f, sq = 0.f;
    #pragma unroll
    for (int k = 0; k < 8; ++k) {
        v[k] = row[lane * 8 + k]; s += v[k]; sq += v[k] * v[k];
    }
    #pragma unroll
    for (int m = 16; m >= 1; m >>= 1) {
        s  += __shfl_xor(s, m, 32);
        sq += __shfl_xor(sq, m, 32);
    }
    float mu = s * (1.0f / CN);
    float ri = rsqrtf(sq * (1.0f / CN) - mu * mu + 1e-5f);
    float* orow = out + t * CN;
    #pragma unroll
    for (int k = 0; k < 8; ++k) {
        int c = lane * 8 + k;
        orow[c] = (v[k] - mu) * ri * g[c] + be[c];
    }
}

// -------------------------------------------------------------------------
// Kernel 8: final 256x256 projection, WMMA f32, transposed store to (B,OUT,L)
// Block = 128 threads = 4 waves; tile = 16 tokens x 256 outputs.
// -------------------------------------------------------------------------
__global__ __launch_bounds__(128) void finalproj_kernel(
    const float* __restrict__ xmn, const float* __restrict__ w,
    const float* __restrict__ pb, float* __restrict__ out)
{
    int t0   = blockIdx.x * 16;
    int wv   = threadIdx.x >> 5;
    int lane = threadIdx.x & 31;
    int row  = lane & 15;
    int kh   = lane >> 4;

    const float* apt = xmn + ((size_t)t0 + row) * CN;
    v8f acc[4] = {};
    for (int kk = 0; kk < CN; kk += 4) {
        __builtin_prefetch(apt + kk + 64, 0, 1);   // global_prefetch_b8
        v2f a; a[0] = apt[kk + 2 * kh]; a[1] = apt[kk + 2 * kh + 1];
        #pragma unroll
        for (int nt = 0; nt < 4; ++nt) {
            int n0 = wv * 64 + nt * 16;
            const float* bpt = w + ((size_t)n0 + row) * CN + kk + 2 * kh;
            v2f b; b[0] = bpt[0]; b[1] = bpt[1];
            acc[nt] = __builtin_amdgcn_wmma_f32_16x16x4_f32(
                false, a, false, b, (short)0, acc[nt], false, false);
        }
    }
    #pragma unroll
    for (int nt = 0; nt < 4; ++nt) {
        int n0 = wv * 64 + nt * 16;
        #pragma unroll
        for (int r = 0; r < 8; ++r) {
            int m = r + 8 * kh, n = n0 + row;
            size_t t = (size_t)t0 + m;
            int bat = (int)(t / Lc), l = (int)(t % Lc);
            out[((size_t)bat * OUTN + n) * Lc + l] = acc[nt][r] + pb[n];
        }
    }
}

// -------------------------------------------------------------------------
extern "C" void kernel_launch(void* const* d_in, const int* in_sizes, int n_in,
                              void* d_out, int out_size, void* d_ws, size_t ws_size,
                              hipStream_t stream) {
    (void)in_sizes; (void)n_in; (void)out_size; (void)ws_size;
    const float* x          = (const float*)d_in[0];
    const float* ln_g       = (const float*)d_in[1];
    const float* ln_b       = (const float*)d_in[2];
    const float* in_proj_w  = (const float*)d_in[3];
    const float* conv_w     = (const float*)d_in[4];
    const float* conv_b     = (const float*)d_in[5];
    const float* x_proj_w   = (const float*)d_in[6];
    const float* dt_proj_w  = (const float*)d_in[7];
    const float* dt_proj_b  = (const float*)d_in[8];
    const float* A_log      = (const float*)d_in[9];
    const float* D_param    = (const float*)d_in[10];
    const float* out_proj_w = (const float*)d_in[11];
    const float* proj_w     = (const float*)d_in[12];
    const float* proj_b     = (const float*)d_in[13];
    const float* skip_scale = (const float*)d_in[14];
    float* out = (float*)d_out;

    // workspace layout (floats)
    float* ws  = (float*)d_ws;
    float* xn  = ws;                               // NT*256      =  2,097,152
    float* xz  = xn  + (size_t)NTc * CN;           // P*NT*256    =  8,388,608
    float* xc  = xz  + (size_t)Pc * NTc * 256;     // P*NT*128    =  4,194,304
    float* dtv = xc  + (size_t)Pc * NTc * DIc;     // P*NT*128    =  4,194,304
    float* bcb = dtv + (size_t)Pc * NTc * DIc;     // P*NT*32     =  1,048,576
    float* xm  = bcb + (size_t)Pc * NTc * 32;      // NT*256      =  2,097,152
    float* xmn = dtv;                              // reuse (dt dead after scan)

    ln1_kernel      <<<Bc * Lc / 32, 256, 0, stream>>>(x, ln_g, ln_b, xn);
    inproj_kernel   <<<dim3(NTc / 16, Pc), 128, 0, stream>>>(xn, in_proj_w, xz);
    conv_kernel     <<<dim3(NTc, Pc), DIc, 0, stream>>>(xz, conv_w, conv_b, xc);
    xproj_kernel    <<<dim3(NTc, Pc), DIc, 0, stream>>>(xc, x_proj_w, dt_proj_w,
                                                        dt_proj_b, dtv, bcb);
    scan_kernel     <<<Bc * Pc, DIc, 0, stream>>>(dtv, bcb, xc, A_log, D_param, xz);
    outproj_kernel  <<<dim3(NTc / 16, Pc), 128, 0, stream>>>(xz, out_proj_w, xn,
                                                             skip_scale, xm);
    ln2_kernel      <<<NTc / 8, 256, 0, stream>>>(xm, ln_g, ln_b, xmn);
    finalproj_kernel<<<NTc / 16, 128, 0, stream>>>(xmn, proj_w, proj_b, out);
}